// HAvgPool1d_72181220376992
// MI455X (gfx1250) — compile-verified
//
#include <hip/hip_runtime.h>
#include <hip/hip_bf16.h>

typedef __attribute__((ext_vector_type(4))) float        v4f;
typedef __attribute__((ext_vector_type(4))) unsigned int v4u;
typedef __attribute__((ext_vector_type(4))) int          v4i;
typedef __attribute__((ext_vector_type(8))) int          v8i;

#define EPSF 1e-7f
#define B_   16
#define C_   256
#define L_   16384
#define OUTL 4096
#define JT   16            // output windows per tile
#define LT   64            // input floats per channel per tile (JT*4)
#define NTHREADS 256
// LDS byte layout (dynamic LDS base == 0, no static __shared__)
#define OFF_IN4   0                         // v4f[256*16]  = 65536 B
#define OFF_OUT   65536                     // float[256*16]= 16384 B
#define OFF_RED4  (65536 + 16384)           // v4f[256]     =  4096 B
#define OFF_A4    (OFF_RED4 + 4096)         // v4f[16]      =   256 B
#define OFF_GJ    (OFF_A4 + 256)            // float[16]    =    64 B
#define OFF_SJ    (OFF_GJ + 64)             // float[16]    =    64 B
#define SMEM_BYTES (OFF_SJ + 64)            // 86400 B

__global__ void __launch_bounds__(NTHREADS)
havgpool1d_kernel(const float* __restrict__ x, float* __restrict__ out) {
    extern __shared__ __align__(16) char smem_raw[];
    v4f*   in4   = (v4f*)(smem_raw + OFF_IN4);    // [c*16 + l4]
    float* outt  = (float*)(smem_raw + OFF_OUT);  // [c*16 + j]
    v4f*   red4  = (v4f*)(smem_raw + OFF_RED4);
    v4f*   aarr4 = (v4f*)(smem_raw + OFF_A4);
    float* Gj    = (float*)(smem_raw + OFF_GJ);
    float* sj    = (float*)(smem_raw + OFF_SJ);

    const unsigned int tid  = threadIdx.x;
    const unsigned int wave = (unsigned int)__builtin_amdgcn_readfirstlane((int)(tid >> 5));
    const unsigned int blk  = blockIdx.x;
    const unsigned int b    = blk >> 8;     // 256 tiles per batch
    const unsigned int jt   = blk & 255u;
    const unsigned int l0   = jt * LT;      // element offset along L
    const unsigned int j0   = jt * JT;      // output offset along out_len

    // ---------- Phase A: TDM async load, one 32-channel slice per wave ----------
    {
        unsigned long long gaddr = (unsigned long long)(uintptr_t)x +
            ((unsigned long long)(b * C_ + wave * 32u) * (unsigned long long)L_ +
             (unsigned long long)l0) * 4ull;

        v4u g0;
        g0[0] = 1u;                                   // count=1, no gather
        g0[1] = wave * 32u * (LT * 4u);               // lds_addr (bytes): wave*8192
        g0[2] = (unsigned int)(gaddr & 0xFFFFFFFFull);
        g0[3] = (unsigned int)((gaddr >> 32) & 0x1FFFFFFull) | (2u << 30); // addr[56:32] | type=2

        v8i g1;
        g1[0] = (int)(2u << 16);                      // data_size = 4 bytes; mask=0
        g1[1] = (int)((L_ & 0xFFFFu) << 16);          // tensor_dim0[15:0] @ bits[63:48]
        g1[2] = (int)(((L_ >> 16) & 0xFFFFu) | (32u << 16));   // dim0 hi | tensor_dim1 lo (=32 ch)
        g1[3] = (int)(((unsigned)LT) << 16);          // tensor_dim1 hi=0 | tile_dim0=64
        g1[4] = (int)(32u);                           // tile_dim1=32, tile_dim2=0
        g1[5] = (int)(L_);                            // tensor_dim0_stride = 16384 elems
        g1[6] = 0;                                    // stride0 hi | stride1 lo
        g1[7] = 0;
        v4i gz;  gz[0] = 0; gz[1] = 0; gz[2] = 0; gz[3] = 0;
        v8i gz8; gz8[0] = 0; gz8[1] = 0; gz8[2] = 0; gz8[3] = 0;
        gz8[4] = 0; gz8[5] = 0; gz8[6] = 0; gz8[7] = 0;

        __builtin_amdgcn_tensor_load_to_lds(g0, g1, gz, gz, gz8, 0);
        __builtin_amdgcn_s_wait_tensorcnt(0);
    }
    __syncthreads();

    // ---------- Phase B: x2[l] = sum_c w^2 (thread = (l4, channel-group)) ----------
    {
        const unsigned int l4 = tid & 15u;
        const unsigned int g  = tid >> 4;
        const unsigned int cbase = g * 16u;
        v4f acc = {0.f, 0.f, 0.f, 0.f};
        #pragma unroll
        for (int c = 0; c < 16; ++c) {
            v4f w = in4[(cbase + c) * 16u + l4];
            acc += w * w;
        }
        red4[tid] = acc;
    }
    __syncthreads();
    if (tid < 16u) {
        v4f x2 = {0.f, 0.f, 0.f, 0.f};
        #pragma unroll
        for (int g = 0; g < 16; ++g) x2 += red4[g * 16 + tid];
        v4f a4; float G = 0.f;
        #pragma unroll
        for (int t = 0; t < 4; ++t) {
            float d   = 1.0f + x2[t];                          // c_curv = 1
            float q   = 1.0f - 4.0f * x2[t] / (d * d);         // 1 - sum_c xK^2
            float gam = rsqrtf(fmaxf(q, EPSF));
            a4[t] = 2.0f * gam / d;                            // gam * (2/(1+x2))
            G    += gam;
        }
        aarr4[tid] = a4;
        Gj[tid]    = G;
    }
    __syncthreads();

    // ---------- Phase C: num[c,j] = sum_t a[t]*w[c,t]; mK2 reduction ----------
    {
        const unsigned int j = tid & 15u;
        const unsigned int g = tid >> 4;
        const unsigned int cbase = g * 16u;
        const v4f a = aarr4[j];
        float p2 = 0.f;
        #pragma unroll
        for (int cc = 0; cc < 16; ++cc) {
            const unsigned int c = cbase + cc;
            v4f w = in4[c * 16u + j];
            float num = a[0]*w[0] + a[1]*w[1] + a[2]*w[2] + a[3]*w[3];
            outt[c * 16u + j] = num;
            p2 += num * num;
        }
        ((float*)red4)[tid] = p2;
    }
    __syncthreads();
    if (tid < 16u) {
        float m2 = 0.f;
        #pragma unroll
        for (int g = 0; g < 16; ++g) m2 += ((float*)red4)[g * 16 + tid];
        const float G   = Gj[tid];
        const float mK2 = m2 / (G * G);
        sj[tid] = 1.0f / (G * (1.0f + sqrtf(fmaxf(1.0f - mK2, EPSF))));
    }
    __syncthreads();

    // ---------- Phase D: scaled coalesced b128 stores ----------
    {
        const unsigned int jj = tid & 3u;     // which float4 of the 16-wide row
        const unsigned int cg = tid >> 2;     // 64 channels per pass
        const v4f sv = ((const v4f*)sj)[jj];
        const v4f* ot4 = (const v4f*)outt;
        #pragma unroll
        for (int p = 0; p < 4; ++p) {
            const unsigned int c = (unsigned int)p * 64u + cg;
            v4f v = ot4[c * 4u + jj] * sv;
            *(v4f*)(out + (unsigned long long)(b * C_ + c) * OUTL + j0 + jj * 4u) = v;
        }
    }
}

extern "C" void kernel_launch(void* const* d_in, const int* in_sizes, int n_in,
                              void* d_out, int out_size, void* d_ws, size_t ws_size,
                              hipStream_t stream) {
    (void)in_sizes; (void)n_in; (void)d_ws; (void)ws_size; (void)out_size;
    const float* x = (const float*)d_in[0];
    float* out = (float*)d_out;
    (void)hipFuncSetAttribute((const void*)havgpool1d_kernel,
                              hipFuncAttributeMaxDynamicSharedMemorySize, SMEM_BYTES);
    dim3 grid(B_ * (OUTL / JT));   // 16 * 256 = 4096 blocks
    dim3 block(NTHREADS);
    havgpool1d_kernel<<<grid, block, SMEM_BYTES, stream>>>(x, out);
}